// PassModel_GAT_52785148068155
// MI455X (gfx1250) — compile-verified
//
#include <hip/hip_runtime.h>
#include <hip/hip_bf16.h>
#include <math.h>

typedef __attribute__((ext_vector_type(2))) float v2f;
typedef __attribute__((ext_vector_type(8))) float v8f;

__device__ __forceinline__ v8f Z8() {
  v8f z = {0.f,0.f,0.f,0.f,0.f,0.f,0.f,0.f};
  return z;
}

// D = A(16x4) * B(4x16) + C, all f32, wave32.
__device__ __forceinline__ v8f wmma4(v2f a, v2f b, v8f c) {
  return __builtin_amdgcn_wmma_f32_16x16x4_f32(false, a, false, b, (short)0, c, false, false);
}

// fast sigmoid: v_exp_f32 + v_rcp_f32 (no IEEE divide expansion)
__device__ __forceinline__ float sigmf(float x) {
  return __builtin_amdgcn_rcpf(1.f + __expf(-x));
}
// fast tanh: CDNA5 v_tanh_f32 if available, else exp+rcp identity
__device__ __forceinline__ float tanh_fast(float x) {
#if __has_builtin(__builtin_amdgcn_tanhf)
  return __builtin_amdgcn_tanhf(x);
#else
  float e = __expf(2.f * x);
  return 1.f - 2.f * __builtin_amdgcn_rcpf(e + 1.f);
#endif
}

// ---- fragment loaders -------------------------------------------------------
// A layout: lanes 0-15: M=lane, VGPR0=K0,VGPR1=K1 ; lanes 16-31: M=lane-16, K2/K3.
// B layout (4x16): VGPR j: lanes0-15 row k0+j, lanes16-31 row k0+j+2; col = lane&15.

// A from a 16x16 row-major LDS matrix (row = M, col = K)
__device__ __forceinline__ v2f ldA16(const float* H, int k0, int lane) {
  int m = lane & 15, h2 = (lane >> 4) * 2;
  v2f a; a.x = H[m * 16 + k0 + h2]; a.y = H[m * 16 + k0 + h2 + 1];
  return a;
}
// B = Wt^T, Wt row-major [outDim, 16]; B[k][n] = Wt[n0+n][k]
__device__ __forceinline__ v2f ldB16(const float* Wt, int n0, int k0, int lane) {
  int n = lane & 15, h2 = (lane >> 4) * 2;
  v2f b; b.x = Wt[(n0 + n) * 16 + k0 + h2]; b.y = Wt[(n0 + n) * 16 + k0 + h2 + 1];
  return b;
}
// B from a zero-padded [48][4] LDS image (layer-0 Wih, K padded 3->4)
__device__ __forceinline__ v2f ldB4(const float* Wt, int n0, int lane) {
  int n = lane & 15, h2 = (lane >> 4) * 2;
  v2f b; b.x = Wt[(n0 + n) * 4 + h2]; b.y = Wt[(n0 + n) * 4 + h2 + 1];
  return b;
}
// B from global row-major weight [outDim, IND]
template<int IND>
__device__ __forceinline__ v2f ldBg(const float* __restrict__ Wt, int n0, int k0, int lane) {
  int n = lane & 15, h2 = (lane >> 4) * 2;
  v2f b;
  b.x = Wt[(size_t)(n0 + n) * IND + k0 + h2];
  b.y = Wt[(size_t)(n0 + n) * IND + k0 + h2 + 1];
  return b;
}

// ---- GRU --------------------------------------------------------------------
struct GruParams {
  const float* Wih[3];
  const float* Whh[3];
  const float* bih[3];
  const float* bhh[3];
};

// LDS weight image offsets (floats):
//  wih0(padded 48x4)=0..192, whh0=192..960, bih0=960, bhh0=1008,
//  wih1=1056, whh1=1824, bih1=2592, bhh1=2640,
//  wih2=2688, whh2=3456, bih2=4224, bhh2=4272, total=4320
#define W_WIH0 0
#define W_WHH0 192
#define W_BIH0 960
#define W_BHH0 1008
#define W_WIH1 1056
#define W_WHH1 1824
#define W_BIH1 2592
#define W_BHH1 2640
#define W_WIH2 2688
#define W_WHH2 3456
#define W_BIH2 4224
#define W_BHH2 4272
#define W_TOT  4320

// One GRU cell update for a 16-node tile. hB: 16x16 LDS state for this layer
// (row-major [node][unit]); inH: previous layer's state (ignored if first).
__device__ __forceinline__ void gru_layer_step(
    bool first, v2f xfrag, const float* inH, float* hB,
    const float* sw, int wihOff, int whhOff, int bihOff, int bhhOff, int lane)
{
  v8f ar = Z8(), az = Z8(), ani = Z8(), anh = Z8();
  if (first) {
    ar  = wmma4(xfrag, ldB4(sw + wihOff, 0,  lane), ar);
    az  = wmma4(xfrag, ldB4(sw + wihOff, 16, lane), az);
    ani = wmma4(xfrag, ldB4(sw + wihOff, 32, lane), ani);
  } else {
#pragma unroll
    for (int c = 0; c < 4; ++c) {
      v2f a = ldA16(inH, 4 * c, lane);
      ar  = wmma4(a, ldB16(sw + wihOff, 0,  4 * c, lane), ar);
      az  = wmma4(a, ldB16(sw + wihOff, 16, 4 * c, lane), az);
      ani = wmma4(a, ldB16(sw + wihOff, 32, 4 * c, lane), ani);
    }
  }
#pragma unroll
  for (int c = 0; c < 4; ++c) {
    v2f a = ldA16(hB, 4 * c, lane);
    ar  = wmma4(a, ldB16(sw + whhOff, 0,  4 * c, lane), ar);
    az  = wmma4(a, ldB16(sw + whhOff, 16, 4 * c, lane), az);
    anh = wmma4(a, ldB16(sw + whhOff, 32, 4 * c, lane), anh);
  }
  const int n = lane & 15, half = lane >> 4;
  const float bir = sw[bihOff + n], biz = sw[bihOff + 16 + n], bin = sw[bihOff + 32 + n];
  const float bhr = sw[bhhOff + n], bhz = sw[bhhOff + 16 + n], bhn = sw[bhhOff + 32 + n];
  float hnew[8];
#pragma unroll
  for (int v = 0; v < 8; ++v) {
    int m = v + 8 * half;
    float r  = sigmf(ar[v] + bir + bhr);
    float z  = sigmf(az[v] + biz + bhz);
    float nn = tanh_fast(ani[v] + bin + r * (anh[v] + bhn));
    hnew[v] = (1.f - z) * nn + z * hB[m * 16 + n];
  }
#pragma unroll
  for (int v = 0; v < 8; ++v) hB[(v + 8 * half) * 16 + n] = hnew[v];
}

// 8 waves/block, each wave owns 16 nodes. All 3 layers pipelined per timestep.
__global__ void __launch_bounds__(256)
gru_stack_kernel(const float* __restrict__ x, GruParams P,
                 float* __restrict__ feat, int nNodes)
{
  __shared__ float sW[W_TOT];
  __shared__ float sH[8][3][256];
  {
    const int t = threadIdx.x, B = blockDim.x;
    // layer-0 Wih, zero-padded K=3 -> 4
    for (int i = t; i < 192; i += B) {
      int nn = i >> 2, k = i & 3;
      sW[W_WIH0 + i] = (k < 3) ? P.Wih[0][nn * 3 + k] : 0.f;
    }
    for (int i = t; i < 768; i += B) {
      sW[W_WIH1 + i] = P.Wih[1][i];
      sW[W_WIH2 + i] = P.Wih[2][i];
      sW[W_WHH0 + i] = P.Whh[0][i];
      sW[W_WHH1 + i] = P.Whh[1][i];
      sW[W_WHH2 + i] = P.Whh[2][i];
    }
    for (int i = t; i < 48; i += B) {
      sW[W_BIH0 + i] = P.bih[0][i]; sW[W_BHH0 + i] = P.bhh[0][i];
      sW[W_BIH1 + i] = P.bih[1][i]; sW[W_BHH1 + i] = P.bhh[1][i];
      sW[W_BIH2 + i] = P.bih[2][i]; sW[W_BHH2 + i] = P.bhh[2][i];
    }
  }
  const int wave = threadIdx.x >> 5, lane = threadIdx.x & 31;
  for (int l = 0; l < 3; ++l)
    for (int i = lane; i < 256; i += 32) sH[wave][l][i] = 0.f;
  __syncthreads();

  const int node0 = (blockIdx.x * 8 + wave) * 16;
  if (node0 >= nNodes) return;                  // uniform per wave, after the only barrier
  int mnode = node0 + (lane & 15);
  if (mnode > nNodes - 1) mnode = nNodes - 1;   // clamp (redundant compute on tail)
  const int half = lane >> 4;
  const float* xbase = x + (size_t)mnode * 150; // [T=50, FEAT=3]

  for (int t = 0; t < 50; ++t) {
    const float* xp = xbase + t * 3;
    // Branchless A-fragment: lanes<16 -> {f0,f1}; lanes>=16 -> {f2,0}
    float x0 = xp[half * 2];
    float x1 = xp[1];
    v2f xf; xf.x = x0; xf.y = half ? 0.f : x1;
    gru_layer_step(true,  xf, sH[wave][0], sH[wave][0], sW, W_WIH0, W_WHH0, W_BIH0, W_BHH0, lane);
    v2f dz; dz.x = 0.f; dz.y = 0.f;
    gru_layer_step(false, dz, sH[wave][0], sH[wave][1], sW, W_WIH1, W_WHH1, W_BIH1, W_BHH1, lane);
    gru_layer_step(false, dz, sH[wave][1], sH[wave][2], sW, W_WIH2, W_WHH2, W_BIH2, W_BHH2, lane);
  }
  // feat[node, l*16+u]  (layer-major concat == transpose(hn,(1,0,2)).reshape)
  for (int idx = lane; idx < 3 * 256; idx += 32) {
    int l = idx >> 8, rem = idx & 255, m = rem >> 4, u = rem & 15;
    int nd = node0 + m;
    if (nd < nNodes) feat[(size_t)nd * 48 + l * 16 + u] = sH[wave][l][rem];
  }
}

// ---- dense projection: out[rows, OUTD] = in[rows, IND] @ Wt[OUTD, IND]^T ----
template<int IND, int OUTD>
__global__ void __launch_bounds__(256)
tile_gemm_kernel(const float* __restrict__ in, const float* __restrict__ Wt,
                 float* __restrict__ out, int nrows)
{
  const int wave = threadIdx.x >> 5, lane = threadIdx.x & 31;
  const int row0 = (blockIdx.x * 8 + wave) * 16;
  if (row0 >= nrows) return;
  int mrow = row0 + (lane & 15);
  if (mrow > nrows - 1) mrow = nrows - 1;
  const int h2 = (lane >> 4) * 2;
  constexpr int CH = IND / 4;
  v2f a[CH];
#pragma unroll
  for (int c = 0; c < CH; ++c) {
    a[c].x = in[(size_t)mrow * IND + 4 * c + h2];
    a[c].y = in[(size_t)mrow * IND + 4 * c + h2 + 1];
  }
  const int n = lane & 15, half = lane >> 4;
#pragma unroll
  for (int ct = 0; ct < OUTD / 16; ++ct) {
    v8f acc = Z8();
#pragma unroll
    for (int c = 0; c < CH; ++c)
      acc = wmma4(a[c], ldBg<IND>(Wt, ct * 16, 4 * c, lane), acc);
#pragma unroll
    for (int v = 0; v < 8; ++v) {
      int r = row0 + v + 8 * half;
      if (r < nrows) out[(size_t)r * OUTD + ct * 16 + n] = acc[v];
    }
  }
}

// ---- GAT edge attention + segment-sum (L2-resident gathers, f32 atomics) ----
__global__ void __launch_bounds__(256)
gat_edge_kernel(const float* __restrict__ h, const float* __restrict__ Wa,
                const int* __restrict__ src, const int* __restrict__ dst,
                float* __restrict__ out, int nE)
{
  __shared__ float sWa[128];
  for (int i = threadIdx.x; i < 128; i += blockDim.x) sWa[i] = Wa[i];
  __syncthreads();
  int e = blockIdx.x * blockDim.x + threadIdx.x;
  if (e >= nE) return;
  const int s = src[e], d = dst[e];
  const float4* hs = (const float4*)(h + (size_t)s * 64);
  const float4* hd = (const float4*)(h + (size_t)d * 64);
  float acc = 0.f;
#pragma unroll
  for (int i = 0; i < 16; ++i) {
    float4 a = hs[i], b = hd[i];
    acc += a.x * sWa[4 * i] + a.y * sWa[4 * i + 1] + a.z * sWa[4 * i + 2] + a.w * sWa[4 * i + 3];
    acc += b.x * sWa[64 + 4 * i] + b.y * sWa[64 + 4 * i + 1] + b.z * sWa[64 + 4 * i + 2] + b.w * sWa[64 + 4 * i + 3];
  }
  const float ec = sigmf(acc);
  float* ob = out + (size_t)d * 64;
#pragma unroll
  for (int i = 0; i < 16; ++i) {
    float4 a = hs[i];
    atomicAdd(ob + 4 * i + 0, ec * a.x);
    atomicAdd(ob + 4 * i + 1, ec * a.y);
    atomicAdd(ob + 4 * i + 2, ec * a.z);
    atomicAdd(ob + 4 * i + 3, ec * a.w);
  }
}

__global__ void zero_kernel(float* __restrict__ p, int n) {
  for (int i = blockIdx.x * blockDim.x + threadIdx.x; i < n; i += gridDim.x * blockDim.x)
    p[i] = 0.f;
}
__global__ void elu_kernel(float* __restrict__ p, int n) {
  for (int i = blockIdx.x * blockDim.x + threadIdx.x; i < n; i += gridDim.x * blockDim.x) {
    float v = p[i];
    p[i] = v > 0.f ? v : (__expf(v) - 1.f);
  }
}

// ---- predictor: 160 -> 256 (ReLU) -> 1 (sigmoid), 16 queries per wave ------
__global__ void __launch_bounds__(128)
predictor_kernel(const float* __restrict__ g, const float* __restrict__ ta,
                 const int* __restrict__ qfrom, const int* __restrict__ qto,
                 const float* __restrict__ W1, const float* __restrict__ b1,
                 const float* __restrict__ W2, const float* __restrict__ b2,
                 float* __restrict__ out, int Q)
{
  __shared__ float sU[4][16 * 160];
  __shared__ float sb1[256];
  __shared__ float sW2[256];
  for (int i = threadIdx.x; i < 256; i += blockDim.x) { sb1[i] = b1[i]; sW2[i] = W2[i]; }
  const int wave = threadIdx.x >> 5, lane = threadIdx.x & 31;
  const int qbase = (blockIdx.x * 4 + wave) * 16;
  float* U = sU[wave];
  for (int idx = lane; idx < 16 * 160; idx += 32) {
    int row = idx / 160, c = idx - row * 160;
    int q = qbase + row; if (q > Q - 1) q = Q - 1;
    int qf = qfrom[q], qt = qto[q];
    float v;
    if      (c < 64)  v = g[(size_t)qf * 64 + c];
    else if (c < 128) v = g[(size_t)qt * 64 + (c - 64)];
    else if (c < 144) v = ta[(size_t)qf * 48 + 32 + (c - 128)];
    else              v = ta[(size_t)qt * 48 + 32 + (c - 144)];
    U[row * 160 + c] = v;
  }
  __syncthreads();

  const int ml = lane & 15, half = lane >> 4, h2 = half * 2;
  float pacc[8];
#pragma unroll
  for (int v = 0; v < 8; ++v) pacc[v] = 0.f;

  for (int ct = 0; ct < 16; ++ct) {
    v8f acc = Z8();
#pragma unroll
    for (int c = 0; c < 40; ++c) {
      const int k0 = 4 * c;
      v2f a; a.x = U[ml * 160 + k0 + h2]; a.y = U[ml * 160 + k0 + h2 + 1];
      v2f b;
      b.x = W1[(size_t)(ct * 16 + ml) * 160 + k0 + h2];
      b.y = W1[(size_t)(ct * 16 + ml) * 160 + k0 + h2 + 1];
      acc = wmma4(a, b, acc);
    }
    const float bb = sb1[ct * 16 + ml];
    const float w2 = sW2[ct * 16 + ml];
#pragma unroll
    for (int v = 0; v < 8; ++v) {
      float hv = acc[v] + bb;
      hv = hv > 0.f ? hv : 0.f;
      pacc[v] += hv * w2;
    }
  }
  // reduce over the 16 lanes of each half (xor masks stay within a half)
#pragma unroll
  for (int off = 1; off < 16; off <<= 1)
#pragma unroll
    for (int v = 0; v < 8; ++v) pacc[v] += __shfl_xor(pacc[v], off, 32);

  if (ml == 0) {
    const float bias2 = b2[0];
#pragma unroll
    for (int v = 0; v < 8; ++v) {
      int q = qbase + v + 8 * half;
      if (q < Q) out[q] = sigmf(pacc[v] + bias2);
    }
  }
}

// ---- host orchestration -----------------------------------------------------
extern "C" void kernel_launch(void* const* d_in, const int* in_sizes, int n_in,
                              void* d_out, int out_size, void* d_ws, size_t ws_size,
                              hipStream_t stream) {
  (void)n_in; (void)out_size; (void)ws_size;
  const float* x    = (const float*)d_in[0];
  const int*   esrc = (const int*)d_in[1];
  const int*   edst = (const int*)d_in[2];
  const int*   qf   = (const int*)d_in[3];
  const int*   qt   = (const int*)d_in[4];

  GruParams na, ta;
  for (int l = 0; l < 3; ++l) {
    na.Wih[l] = (const float*)d_in[5 + 4 * l];
    na.Whh[l] = (const float*)d_in[6 + 4 * l];
    na.bih[l] = (const float*)d_in[7 + 4 * l];
    na.bhh[l] = (const float*)d_in[8 + 4 * l];
    ta.Wih[l] = (const float*)d_in[17 + 4 * l];
    ta.Whh[l] = (const float*)d_in[18 + 4 * l];
    ta.bih[l] = (const float*)d_in[19 + 4 * l];
    ta.bhh[l] = (const float*)d_in[20 + 4 * l];
  }
  const float* gW1  = (const float*)d_in[29]; // [64, 48]
  const float* gWa1 = (const float*)d_in[30]; // [128]
  const float* gW2  = (const float*)d_in[31]; // [64, 64]
  const float* gWa2 = (const float*)d_in[32]; // [128]
  const float* pW1  = (const float*)d_in[33]; // [256, 160]
  const float* pb1  = (const float*)d_in[34]; // [256]
  const float* pW2  = (const float*)d_in[35]; // [1, 256]
  const float* pb2  = (const float*)d_in[36]; // [1]

  const int N = in_sizes[0] / 150;   // [N, 50, 3]
  const int E = in_sizes[1];
  const int Q = in_sizes[3];

  float* ws      = (float*)d_ws;
  float* feat_na = ws;                        // [N, 48]
  float* feat_ta = ws + (size_t)N * 48;       // [N, 48]
  float* hbuf    = ws + (size_t)N * 96;       // [N, 64]
  float* gbuf    = ws + (size_t)N * 160;      // [N, 64]

  const int tileBlocks = (N + 127) / 128;     // 8 waves x 16 rows per block

  gru_stack_kernel<<<tileBlocks, 256, 0, stream>>>(x, na, feat_na, N);
  gru_stack_kernel<<<tileBlocks, 256, 0, stream>>>(x, ta, feat_ta, N);

  tile_gemm_kernel<48, 64><<<tileBlocks, 256, 0, stream>>>(feat_na, gW1, hbuf, N);
  zero_kernel<<<1024, 256, 0, stream>>>(gbuf, N * 64);
  gat_edge_kernel<<<(E + 255) / 256, 256, 0, stream>>>(hbuf, gWa1, esrc, edst, gbuf, E);
  elu_kernel<<<1024, 256, 0, stream>>>(gbuf, N * 64);

  tile_gemm_kernel<64, 64><<<tileBlocks, 256, 0, stream>>>(gbuf, gW2, hbuf, N);
  zero_kernel<<<1024, 256, 0, stream>>>(gbuf, N * 64);
  gat_edge_kernel<<<(E + 255) / 256, 256, 0, stream>>>(hbuf, gWa2, esrc, edst, gbuf, E);

  predictor_kernel<<<(Q + 63) / 64, 128, 0, stream>>>(gbuf, feat_ta, qf, qt,
                                                      pW1, pb1, pW2, pb2,
                                                      (float*)d_out, Q);
}